// LocalGaussianBlur3D_64123861729376
// MI455X (gfx1250) — compile-verified
//
#include <hip/hip_runtime.h>

#define RAD   4
#define KS    9          // 2R+1
#define EMAX  17         // max extended extent: 9 + 2R
#define OMAX  9          // max box extent
#define DIMV  256
#define SIGMA 1.2f

typedef __attribute__((ext_vector_type(2))) float v2f;
typedef __attribute__((ext_vector_type(8))) float v8f;

// ---------------- Kernel 1: streaming copy volume -> out (HBM-bound) -------
__global__ void lgb_copy_kernel(const float4* __restrict__ src,
                                float4* __restrict__ dst, int n4) {
    int stride = gridDim.x * blockDim.x;
    for (int i = blockIdx.x * blockDim.x + threadIdx.x; i < n4; i += stride)
        dst[i] = src[i];
}

__device__ __forceinline__ int iabs_(int a) { return a < 0 ? -a : a; }

// ---------------- Kernel 2: per-point box blur, x-pass via WMMA f32 --------
__global__ __launch_bounds__(256, 1)
void lgb_blur_boxes(const float* __restrict__ vol, const int* __restrict__ pts,
                    float* __restrict__ out) {
    __shared__ float Lds[EMAX * EMAX * EMAX];  // zero-padded extended input
    __shared__ float T1 [EMAX * EMAX * 16];    // after x-pass (cols padded to 16)
    __shared__ float T2 [EMAX * OMAX * 16];    // after y-pass
    __shared__ float gw [KS];                  // normalized gaussian weights

    const int tid   = threadIdx.x;
    const int lane  = tid & 31;
    const int wave  = tid >> 5;
    const int nwav  = blockDim.x >> 5;         // 8 waves
    const int pi    = blockIdx.x;              // point index

    const int pz = pts[3 * pi + 0], py = pts[3 * pi + 1], px = pts[3 * pi + 2];
    const int z0 = max(0, pz - RAD), z1 = min(DIMV - 1, pz + RAD);
    const int y0 = max(0, py - RAD), y1 = min(DIMV - 1, py + RAD);
    const int x0 = max(0, px - RAD), x1 = min(DIMV - 1, px + RAD);
    const int Oz = z1 - z0 + 1, Oy = y1 - y0 + 1, Ox = x1 - x0 + 1;
    const int Ez = Oz + 2 * RAD, Ey = Oy + 2 * RAD, Ex = Ox + 2 * RAD;

    // Gaussian weights, matching reference f32 math (exp, normalize, 1e-12 floor)
    if (tid == 0) {
        float tmp[KS], s = 0.f;
        for (int i = 0; i < KS; ++i) {
            float x = (float)i - (float)(KS - 1) * 0.5f;
            tmp[i] = expf(-(x * x) / (2.f * SIGMA * SIGMA));
            s += tmp[i];
        }
        s = fmaxf(s, 1e-12f);
        for (int i = 0; i < KS; ++i) gw[i] = tmp[i] / s;
    }

    // Load extended region into LDS; zeros outside the volume == conv zero padding
    const int Etot = Ez * Ey * Ex;
    for (int idx = tid; idx < Etot; idx += blockDim.x) {
        int ex = idx % Ex, t = idx / Ex;
        int ey = t % Ey,  ez = t / Ey;
        int Z = z0 - RAD + ez, Y = y0 - RAD + ey, X = x0 - RAD + ex;
        float v = 0.f;
        if (Z >= 0 && Z < DIMV && Y >= 0 && Y < DIMV && X >= 0 && X < DIMV)
            v = vol[((size_t)Z * DIMV + Y) * DIMV + X];
        Lds[(ez * EMAX + ey) * EMAX + ex] = v;
    }
    __syncthreads();

    // -------- Pass 1 (x): T1[z][y][j] = sum_c gw[c]*L[z][y][j+c] via WMMA ----
    // D(16y x 16j) = A(16y x K) * B(K x 16j), B[k][j] = gw[k-j] (banded)
    const int mtiles = (Ey + 15) >> 4;   // 1..2
    const int ksteps = (Ex + 3) >> 2;    // <=5, K padded with zeros
    const int half   = lane >> 4;        // K half select (ISA f32 A/B layout)
    const int nl     = lane & 15;        // M (rows) for A, N (cols) for B/D
    for (int z = wave; z < Ez; z += nwav) {
        for (int mt = 0; mt < mtiles; ++mt) {
            v8f acc = {};
            const int row = mt * 16 + nl;
            const bool rok = row < Ey;
            for (int ks = 0; ks < ksteps; ++ks) {
                int kb = ks * 4 + half * 2;  // lanes 0-15: K={0,1}; 16-31: K={2,3}
                v2f a, b;
                a.x = (rok && kb     < Ex) ? Lds[(z * EMAX + row) * EMAX + kb    ] : 0.f;
                a.y = (rok && kb + 1 < Ex) ? Lds[(z * EMAX + row) * EMAX + kb + 1] : 0.f;
                int d0 = kb - nl, d1 = kb + 1 - nl;
                b.x = (d0 >= 0 && d0 < KS) ? gw[d0] : 0.f;
                b.y = (d1 >= 0 && d1 < KS) ? gw[d1] : 0.f;
                acc = __builtin_amdgcn_wmma_f32_16x16x4_f32(
                    false, a, false, b, (short)0, acc, false, false);
            }
            // D layout: VGPR v -> row v (lanes 0-15) / v+8 (lanes 16-31), col = nl
#pragma unroll
            for (int v = 0; v < 8; ++v) {
                int r = mt * 16 + v + half * 8;
                if (r < Ey) T1[(z * EMAX + r) * 16 + nl] = acc[v];
            }
        }
    }
    __syncthreads();

    // -------- Pass 2 (y): T2[z][i][j] = sum_b gw[b]*T1[z][i+b][j] -----------
    const int P2 = Ez * Oy * Ox;
    for (int idx = tid; idx < P2; idx += blockDim.x) {
        int j = idx % Ox, t = idx / Ox;
        int i = t % Oy,  z = t / Oy;
        float s = 0.f;
#pragma unroll
        for (int b = 0; b < KS; ++b)
            s += gw[b] * T1[(z * EMAX + (i + b)) * 16 + j];
        T2[(z * OMAX + i) * 16 + j] = s;
    }
    __syncthreads();

    // -------- Pass 3 (z) + ownership-filtered global write ------------------
    const int P3 = Oz * Oy * Ox;
    for (int idx = tid; idx < P3; idx += blockDim.x) {
        int j = idx % Ox, t = idx / Ox;
        int i = t % Oy,  a = t / Oy;
        float s = 0.f;
#pragma unroll
        for (int c = 0; c < KS; ++c)
            s += gw[c] * T2[((a + c) * OMAX + i) * 16 + j];
        int Z = z0 + a, Y = y0 + i, X = x0 + j;
        // only the lowest-index point containing this voxel writes it (no races)
        bool owned = true;
        for (int q = 0; q < pi; ++q) {
            if (iabs_(Z - pts[3 * q + 0]) <= RAD &&
                iabs_(Y - pts[3 * q + 1]) <= RAD &&
                iabs_(X - pts[3 * q + 2]) <= RAD) { owned = false; break; }
        }
        if (owned) out[((size_t)Z * DIMV + Y) * DIMV + X] = s;
    }
}

extern "C" void kernel_launch(void* const* d_in, const int* in_sizes, int n_in,
                              void* d_out, int out_size, void* d_ws, size_t ws_size,
                              hipStream_t stream) {
    const float* vol = (const float*)d_in[0];
    const int*   pts = (const int*)d_in[1];
    float*       out = (float*)d_out;

    const int n4   = out_size / 4;        // 256^3 / 4 float4 elements
    const int npts = in_sizes[1] / 3;     // 6

    lgb_copy_kernel<<<4096, 256, 0, stream>>>((const float4*)vol, (float4*)out, n4);
    lgb_blur_boxes<<<npts, 256, 0, stream>>>(vol, pts, out);
}